// DecoderRNN_48241072668944
// MI455X (gfx1250) — compile-verified
//
#include <hip/hip_runtime.h>

// ---------------------------------------------------------------------------
// DecoderRNN single step on gfx1250 (MI455X).
// Bandwidth-bound GEMV chain (~244 MB weights/call -> ~10.5 us at 23.3 TB/s).
// WMMA f32 16x16x4 does the accumulation on the matrix pipe; weights stream
// f32 non-temporal; activations staged in LDS (real reuse). Dual accumulators
// break the WMMA C->D dependence chain; no per-step A masking (unused D rows
// may hold duplicates, only rows 0/1 are read).
// ---------------------------------------------------------------------------

typedef __attribute__((ext_vector_type(2))) float v2f;
typedef __attribute__((ext_vector_type(8))) float v8f;

#define HID   1024
#define HP    1028          // H + 4
#define VOCAB 50257

static __device__ __forceinline__ v8f wmma4(v2f a, v2f b, v8f c) {
  // (neg_a, A, neg_b, B, c_mod, C, reuse_a, reuse_b)
  return __builtin_amdgcn_wmma_f32_16x16x4_f32(false, a, false, b, (short)0, c,
                                               false, false);
}

static __device__ __forceinline__ v2f ntload2(const float* p) {
  return __builtin_nontemporal_load(reinterpret_cast<const v2f*>(p));
}

// --- K1: reformat h,c (shared W_ref read once: hidden -> A row0, cell -> A
//         row1 via per-lane LDS base select) + embed/relu in spare block.
__global__ void k1_reformat_embed(const int* __restrict__ tok,
                                  const float* __restrict__ hidden,
                                  const float* __restrict__ cell,
                                  const float* __restrict__ W_ref,
                                  const float* __restrict__ b_ref,
                                  const float* __restrict__ emb,
                                  float* __restrict__ x_out,
                                  float* __restrict__ hproj,
                                  float* __restrict__ cproj) {
  if (blockIdx.x == 8) {                       // embed + relu branch
    const int t = tok[0];
    for (int j = threadIdx.x; j < HID; j += 256) {
      float v = emb[(long)t * HID + j];
      x_out[j] = v > 0.f ? v : 0.f;
    }
    return;
  }
  __shared__ float sh[HP];
  __shared__ float sc[HP];
  for (int j = threadIdx.x; j < HP; j += 256) { sh[j] = hidden[j]; sc[j] = cell[j]; }
  __syncthreads();

  const int wave  = threadIdx.x >> 5;
  const int lane  = threadIdx.x & 31;
  const int tile  = blockIdx.x * 8 + wave;     // 0..63  (64*16 = 1024 rows)
  const int row16 = lane & 15;
  const int koff  = (lane < 16) ? 0 : 2;
  const int orow  = tile * 16 + row16;
  const float* wrow = W_ref + (long)orow * HP;
  // A-row select: M=0 must be hidden, M=1 must be cell, M>=2 don't-care.
  const float* av = (row16 == 1) ? sc : sh;

  v8f acc0 = {}, acc1 = {};
  for (int k = 0; k < HID; k += 8) {           // 1024 of 1028, dual-acc
    v2f b0 = ntload2(wrow + k + koff);
    v2f b1 = ntload2(wrow + k + 4 + koff);
    v2f a0 = *reinterpret_cast<const v2f*>(av + k + koff);
    v2f a1 = *reinterpret_cast<const v2f*>(av + k + 4 + koff);
    acc0 = wmma4(a0, b0, acc0);
    acc1 = wmma4(a1, b1, acc1);
  }
  { // K tail: k = 1024..1027
    v2f b = ntload2(wrow + HID + koff);
    v2f a = *reinterpret_cast<const v2f*>(av + HID + koff);
    acc0 = wmma4(a, b, acc0);
  }
  if (lane < 16) {
    hproj[orow] = acc0[0] + acc1[0] + b_ref[orow];   // D row M=0
    cproj[orow] = acc0[1] + acc1[1] + b_ref[orow];   // D row M=1
  }
}

// --- K2: gates = x*W_ih^T + h*W_hh^T + b_ih + b_hh
//         (independent accumulators for the two weight streams)
__global__ void k2_gates(const float* __restrict__ x,
                         const float* __restrict__ h,
                         const float* __restrict__ W_ih,
                         const float* __restrict__ W_hh,
                         const float* __restrict__ b_ih,
                         const float* __restrict__ b_hh,
                         float* __restrict__ gates) {
  __shared__ float sx[HID];
  __shared__ float sh[HID];
  for (int j = threadIdx.x; j < HID; j += 256) { sx[j] = x[j]; sh[j] = h[j]; }
  __syncthreads();

  const int wave = threadIdx.x >> 5;
  const int lane = threadIdx.x & 31;
  const int tile = blockIdx.x * 8 + wave;      // 0..255 (256*16 = 4096 gates)
  const int koff = (lane < 16) ? 0 : 2;
  const int g    = tile * 16 + (lane & 15);
  const float* wi = W_ih + (long)g * HID;
  const float* wh = W_hh + (long)g * HID;

  v8f acc0 = {}, acc1 = {};
  for (int k = 0; k < HID; k += 4) {
    v2f bi = ntload2(wi + k + koff);
    v2f bh = ntload2(wh + k + koff);
    v2f xv = *reinterpret_cast<const v2f*>(&sx[k + koff]);
    v2f hv = *reinterpret_cast<const v2f*>(&sh[k + koff]);
    acc0 = wmma4(xv, bi, acc0);                // x . W_ih
    acc1 = wmma4(hv, bh, acc1);                // h . W_hh
  }
  if (lane < 16) gates[g] = acc0[0] + acc1[0] + b_ih[g] + b_hh[g];
}

// --- K3: LSTM pointwise (torch gate order i,f,g,o)
__global__ void k3_pointwise(const float* __restrict__ gates,
                             const float* __restrict__ cproj,
                             float* __restrict__ hnew,
                             float* __restrict__ cnew) {
  const int j = blockIdx.x * 256 + threadIdx.x;
  if (j >= HID) return;
  const float ig = gates[j];
  const float fg = gates[HID + j];
  const float gg = gates[2 * HID + j];
  const float og = gates[3 * HID + j];
  const float si = 1.f / (1.f + __expf(-ig));
  const float sf = 1.f / (1.f + __expf(-fg));
  const float so = 1.f / (1.f + __expf(-og));
  const float c  = sf * cproj[j] + si * tanhf(gg);
  cnew[j] = c;
  hnew[j] = so * tanhf(c);
}

// --- K4: logits = h_new * W_out^T + b_out  (dominant 206 MB NT stream)
__global__ void k4_outproj(const float* __restrict__ hnew,
                           const float* __restrict__ W_out,
                           const float* __restrict__ b_out,
                           float* __restrict__ out) {
  __shared__ float sh[HID];
  for (int j = threadIdx.x; j < HID; j += 256) sh[j] = hnew[j];
  __syncthreads();

  const int wave  = threadIdx.x >> 5;
  const int lane  = threadIdx.x & 31;
  const int tile  = blockIdx.x * 8 + wave;
  const int ntile = (VOCAB + 15) / 16;         // 3142
  if (tile >= ntile) return;                   // wave-uniform exit
  const int row16 = lane & 15;
  const int koff  = (lane < 16) ? 0 : 2;
  const int v     = tile * 16 + row16;
  const int vc    = v < VOCAB ? v : VOCAB - 1; // clamp loads, keep EXEC full
  const float* wrow = W_out + (long)vc * HID;

  v8f acc0 = {}, acc1 = {};
  for (int k = 0; k < HID; k += 8) {
    v2f b0 = ntload2(wrow + k + koff);
    v2f b1 = ntload2(wrow + k + 4 + koff);
    v2f a0 = *reinterpret_cast<const v2f*>(&sh[k + koff]);
    v2f a1 = *reinterpret_cast<const v2f*>(&sh[k + 4 + koff]);
    acc0 = wmma4(a0, b0, acc0);
    acc1 = wmma4(a1, b1, acc1);
  }
  if (lane < 16 && v < VOCAB) out[v] = acc0[0] + acc1[0] + b_out[v];
}

// ---------------------------------------------------------------------------
extern "C" void kernel_launch(void* const* d_in, const int* in_sizes, int n_in,
                              void* d_out, int out_size, void* d_ws, size_t ws_size,
                              hipStream_t stream) {
  (void)in_sizes; (void)n_in; (void)out_size; (void)ws_size;

  const int*   tok    = (const int*)  d_in[0];
  const float* hidden = (const float*)d_in[1];
  const float* cell   = (const float*)d_in[2];
  const float* W_ref  = (const float*)d_in[3];
  const float* b_ref  = (const float*)d_in[4];
  const float* emb    = (const float*)d_in[5];
  const float* W_ih   = (const float*)d_in[6];
  const float* W_hh   = (const float*)d_in[7];
  const float* b_ih   = (const float*)d_in[8];
  const float* b_hh   = (const float*)d_in[9];
  const float* W_out  = (const float*)d_in[10];
  const float* b_out  = (const float*)d_in[11];

  float* out  = (float*)d_out;          // [V] logits
  float* hnew = out + VOCAB;            // [H]
  float* cnew = hnew + HID;             // [H]

  float* ws    = (float*)d_ws;
  float* x     = ws;                    // H
  float* hproj = ws + HID;              // H
  float* cproj = ws + 2 * HID;          // H
  float* gates = ws + 3 * HID;          // 4H

  k1_reformat_embed<<<9, 256, 0, stream>>>(tok, hidden, cell, W_ref, b_ref, emb,
                                           x, hproj, cproj);
  k2_gates<<<32, 256, 0, stream>>>(x, hproj, W_ih, W_hh, b_ih, b_hh, gates);
  k3_pointwise<<<4, 256, 0, stream>>>(gates, cproj, hnew, cnew);

  const int ntile  = (VOCAB + 15) / 16;        // 3142 wave-tiles
  const int blocks = (ntile + 7) / 8;          // 8 waves / block -> 393
  k4_outproj<<<blocks, 256, 0, stream>>>(hnew, W_out, b_out, out);
}